// TGN_49185965474385
// MI455X (gfx1250) — compile-verified
//
#include <hip/hip_runtime.h>
#include <stdint.h>

typedef float v4f __attribute__((ext_vector_type(4)));
typedef int   v4i __attribute__((vector_size(16)));
typedef long long i64;

#define TGN_D   172
#define TGN_CH  43          // 172 floats = 43 x 16B chunks per row
#define TGN_B   4096
#define TGN_K   20
#define TGN_N   100000
#define TGN_RPI 4           // rows per async iteration (per wave)

// ---- CDNA5 async global<->LDS DMA path (gfx1250), guarded so we always compile ----
#if defined(__AMDGCN__) && \
    __has_builtin(__builtin_amdgcn_global_load_async_to_lds_b128) && \
    __has_builtin(__builtin_amdgcn_global_store_async_from_lds_b128)
#define TGN_ASYNC 1
#else
#define TGN_ASYNC 0
#endif

#if TGN_ASYNC
typedef __attribute__((address_space(1))) v4i as1_v4i;   // global int4
typedef __attribute__((address_space(3))) v4i as3_v4i;   // LDS int4
// Generic flat pointer -> AS1: flat global address == 64-bit VA.
__device__ __forceinline__ as1_v4i* tgn_as1(const void* p) {
  return (as1_v4i*)(unsigned long long)p;
}
// Generic flat pointer to __shared__ -> AS3: per ISA 10.2, the LDS aperture keeps
// the LDS byte offset in addr[31:0], so truncation yields the LDS address.
__device__ __forceinline__ as3_v4i* tgn_as3(const void* p) {
  return (as3_v4i*)(unsigned int)(unsigned long long)p;
}
__device__ __forceinline__ void tgn_wait_async0() {
#if __has_builtin(__builtin_amdgcn_s_wait_asynccnt)
  __builtin_amdgcn_s_wait_asynccnt(0);
#else
  asm volatile("s_wait_asynccnt 0" ::: "memory");
#endif
}
#endif

// ==== pure-copy gather, single output: CDNA5 async LDS-DMA (first in file) ====
__global__ __launch_bounds__(256) void tgn_gather_copy(const v4f* __restrict__ srcA,
                                                       const int* __restrict__ idx,
                                                       v4f* __restrict__ out0,
                                                       unsigned nrows) {
  const unsigned lane = threadIdx.x & 31u;
  const unsigned wid  = blockIdx.x * 8u + (threadIdx.x >> 5);
  const unsigned wtot = gridDim.x * 8u;
#if TGN_ASYNC
  __shared__ char slab_all[8][TGN_RPI * TGN_CH * 16];   // private slab per wave, no barriers
  char* slab = slab_all[threadIdx.x >> 5];
  for (unsigned g = wid; g * TGN_RPI < nrows; g += wtot) {
    const unsigned r0 = g * TGN_RPI;
    const unsigned nr = (nrows - r0) < TGN_RPI ? (nrows - r0) : TGN_RPI;
    for (unsigned j = 0; j < nr; ++j) {               // stage rows into LDS via DMA
      const i64 s = (i64)idx[r0 + j];
      const char* gsrc = (const char*)(srcA + s * TGN_CH);
      char* l = slab + j * (TGN_CH * 16);
      __builtin_amdgcn_global_load_async_to_lds_b128(tgn_as1(gsrc + lane * 16),
                                                     tgn_as3(l + lane * 16), 0, 0);
      if (lane < TGN_CH - 32)
        __builtin_amdgcn_global_load_async_to_lds_b128(tgn_as1(gsrc + (32 + lane) * 16),
                                                       tgn_as3(l + (32 + lane) * 16), 0, 0);
    }
    tgn_wait_async0();
    for (unsigned j = 0; j < nr; ++j) {               // stream rows out of LDS via DMA
      char* l = slab + j * (TGN_CH * 16);
      char* o = (char*)(out0 + (i64)(r0 + j) * TGN_CH);
      __builtin_amdgcn_global_store_async_from_lds_b128(tgn_as1(o + lane * 16),
                                                        tgn_as3(l + lane * 16), 0, 0);
      if (lane < TGN_CH - 32)
        __builtin_amdgcn_global_store_async_from_lds_b128(tgn_as1(o + (32 + lane) * 16),
                                                          tgn_as3(l + (32 + lane) * 16), 0, 0);
    }
    tgn_wait_async0();   // LDS slab reused next iteration
  }
#else
  for (unsigned r = wid; r < nrows; r += wtot) {
    const i64 s = (i64)idx[r];
    const v4f* sp = srcA + s * TGN_CH;
    v4f* op = out0 + (i64)r * TGN_CH;
    __builtin_nontemporal_store(sp[lane], &op[lane]);
    if (lane < TGN_CH - 32)
      __builtin_nontemporal_store(sp[32 + lane], &op[32 + lane]);
  }
#endif
}

// ==== pure-copy gather, dual output (src_feats2 == neigh_feats1 reshaped) ====
__global__ __launch_bounds__(256) void tgn_gather_copy2(const v4f* __restrict__ srcA,
                                                        const int* __restrict__ idx,
                                                        v4f* __restrict__ out0,
                                                        v4f* __restrict__ out1,
                                                        unsigned nrows) {
  const unsigned lane = threadIdx.x & 31u;
  const unsigned wid  = blockIdx.x * 8u + (threadIdx.x >> 5);
  const unsigned wtot = gridDim.x * 8u;
#if TGN_ASYNC
  __shared__ char slab_all[8][TGN_RPI * TGN_CH * 16];
  char* slab = slab_all[threadIdx.x >> 5];
  for (unsigned g = wid; g * TGN_RPI < nrows; g += wtot) {
    const unsigned r0 = g * TGN_RPI;
    const unsigned nr = (nrows - r0) < TGN_RPI ? (nrows - r0) : TGN_RPI;
    for (unsigned j = 0; j < nr; ++j) {
      const i64 s = (i64)idx[r0 + j];
      const char* gsrc = (const char*)(srcA + s * TGN_CH);
      char* l = slab + j * (TGN_CH * 16);
      __builtin_amdgcn_global_load_async_to_lds_b128(tgn_as1(gsrc + lane * 16),
                                                     tgn_as3(l + lane * 16), 0, 0);
      if (lane < TGN_CH - 32)
        __builtin_amdgcn_global_load_async_to_lds_b128(tgn_as1(gsrc + (32 + lane) * 16),
                                                       tgn_as3(l + (32 + lane) * 16), 0, 0);
    }
    tgn_wait_async0();
    for (unsigned j = 0; j < nr; ++j) {
      char* l = slab + j * (TGN_CH * 16);
      char* o  = (char*)(out0 + (i64)(r0 + j) * TGN_CH);
      char* o2 = (char*)(out1 + (i64)(r0 + j) * TGN_CH);
      __builtin_amdgcn_global_store_async_from_lds_b128(tgn_as1(o + lane * 16),
                                                        tgn_as3(l + lane * 16), 0, 0);
      __builtin_amdgcn_global_store_async_from_lds_b128(tgn_as1(o2 + lane * 16),
                                                        tgn_as3(l + lane * 16), 0, 0);
      if (lane < TGN_CH - 32) {
        __builtin_amdgcn_global_store_async_from_lds_b128(tgn_as1(o + (32 + lane) * 16),
                                                          tgn_as3(l + (32 + lane) * 16), 0, 0);
        __builtin_amdgcn_global_store_async_from_lds_b128(tgn_as1(o2 + (32 + lane) * 16),
                                                          tgn_as3(l + (32 + lane) * 16), 0, 0);
      }
    }
    tgn_wait_async0();
  }
#else
  for (unsigned r = wid; r < nrows; r += wtot) {
    const i64 s = (i64)idx[r];
    const v4f* sp = srcA + s * TGN_CH;
    v4f* op0 = out0 + (i64)r * TGN_CH;
    v4f* op1 = out1 + (i64)r * TGN_CH;
    v4f v0 = sp[lane];
    __builtin_nontemporal_store(v0, &op0[lane]);
    __builtin_nontemporal_store(v0, &op1[lane]);
    if (lane < TGN_CH - 32) {
      v4f v1 = sp[32 + lane];
      __builtin_nontemporal_store(v1, &op0[32 + lane]);
      __builtin_nontemporal_store(v1, &op1[32 + lane]);
    }
  }
#endif
}

// ==== fused (a+b) gather: fallback when workspace unavailable ====
__global__ __launch_bounds__(256) void tgn_gather_add(const v4f* __restrict__ srcA,
                                                      const v4f* __restrict__ srcB,
                                                      const int* __restrict__ idx,
                                                      v4f* __restrict__ out0,
                                                      v4f* __restrict__ out1,   // nullable
                                                      unsigned nrows) {
  const unsigned lane = threadIdx.x & 31u;
  const unsigned wid  = blockIdx.x * 8u + (threadIdx.x >> 5);
  const unsigned wtot = gridDim.x * 8u;
  for (unsigned r = wid; r < nrows; r += wtot) {
    const i64 s = (i64)idx[r];
    const v4f* pa = srcA + s * TGN_CH;
    const v4f* pb = srcB + s * TGN_CH;
    v4f* op = out0 + (i64)r * TGN_CH;
    v4f v0 = pa[lane] + pb[lane];
    __builtin_nontemporal_store(v0, &op[lane]);
    if (out1) __builtin_nontemporal_store(v0, &(out1 + (i64)r * TGN_CH)[lane]);
    if (lane < TGN_CH - 32) {
      v4f v1 = pa[32 + lane] + pb[32 + lane];
      __builtin_nontemporal_store(v1, &op[32 + lane]);
      if (out1) __builtin_nontemporal_store(v1, &(out1 + (i64)r * TGN_CH)[32 + lane]);
    }
  }
}

// ---------------- precompute node_mem = node_feats + memory (L2-resident table) ---------------
__global__ __launch_bounds__(256) void tgn_add_kernel(const v4f* __restrict__ a,
                                                      const v4f* __restrict__ b,
                                                      v4f* __restrict__ o, unsigned n4) {
  unsigned i = blockIdx.x * blockDim.x + threadIdx.x;
  unsigned stride = gridDim.x * blockDim.x;
  for (; i < n4; i += stride) {
    o[i] = a[i] + b[i];   // regular (temporal) store: we WANT this hot in L2
  }
}

// ---------------- deltas + masks (elementwise, tiny; 32-bit index math) ----------------
__global__ __launch_bounds__(256) void tgn_delta_mask_kernel(const float* __restrict__ ts,
                                                             const float* __restrict__ et,
                                                             const int* __restrict__ nbr,
                                                             float* __restrict__ deltas,
                                                             float* __restrict__ mask,
                                                             unsigned n, unsigned divi) {
  unsigned i = blockIdx.x * blockDim.x + threadIdx.x;
  unsigned stride = gridDim.x * blockDim.x;
  for (; i < n; i += stride) {
    const float t = ts[i / divi];
    __builtin_nontemporal_store(t - et[i], &deltas[i]);
    __builtin_nontemporal_store(nbr[i] == 0 ? 1.0f : 0.0f, &mask[i]);
  }
}

extern "C" void kernel_launch(void* const* d_in, const int* in_sizes, int n_in,
                              void* d_out, int out_size, void* d_ws, size_t ws_size,
                              hipStream_t stream) {
  (void)in_sizes; (void)n_in; (void)out_size;
  const float* node_feats  = (const float*)d_in[0];
  const float* edge_feats  = (const float*)d_in[1];
  const float* memory      = (const float*)d_in[2];
  const float* timestamps  = (const float*)d_in[3];
  const float* edge_times1 = (const float*)d_in[4];
  const float* edge_times2 = (const float*)d_in[5];
  const int*   source_nodes= (const int*)d_in[6];
  const int*   neighbors1  = (const int*)d_in[7];
  const int*   edge_idxs1  = (const int*)d_in[8];
  const int*   neighbors2  = (const int*)d_in[9];
  const int*   edge_idxs2  = (const int*)d_in[10];

  const i64 B = TGN_B, K = TGN_K, D = TGN_D, N = TGN_N;
  float* out = (float*)d_out;

  // flat output offsets (reference return order)
  const i64 o0 = 0;                       // src_feats1   [B, D]
  const i64 o1 = o0 + B * D;              // neigh_feats1 [B*K, D]
  const i64 o2 = o1 + B * K * D;          // edge_f1      [B*K, D]
  const i64 o3 = o2 + B * K * D;          // deltas1      [B*K]
  const i64 o4 = o3 + B * K;              // mask1        [B*K]
  const i64 o5 = o4 + B * K;              // src_feats2   [B*K, D]  (== neigh_feats1 data)
  const i64 o6 = o5 + B * K * D;          // neigh_feats2 [B*K*K, D]
  const i64 o7 = o6 + B * K * K * D;      // edge_f2      [B*K*K, D]
  const i64 o8 = o7 + B * K * K * D;      // deltas2      [B*K*K]
  const i64 o9 = o8 + B * K * K;          // mask2        [B*K*K]

  // Precompute node_mem into workspace if it fits (68.8 MB -> stays hot in 192 MB L2)
  const bool use_ws = ws_size >= (size_t)(N * D) * sizeof(float);
  if (use_ws) {
    const unsigned n4 = (unsigned)(N * D / 4);
    tgn_add_kernel<<<dim3((n4 + 255) / 256), dim3(256), 0, stream>>>(
        (const v4f*)node_feats, (const v4f*)memory, (v4f*)d_ws, n4);
  }
  const float* nodeA = use_ws ? (const float*)d_ws : node_feats;

  auto copy_blocks = [](i64 rows) -> unsigned {
    i64 waves = (rows + TGN_RPI - 1) / TGN_RPI;
    i64 blocks = (waves + 7) / 8;          // 8 waves per 256-thread block
    return (unsigned)(blocks < 1 ? 1 : blocks);
  };
  auto add_blocks = [](i64 rows) -> unsigned {
    i64 blocks = (rows + 7) / 8;
    return (unsigned)(blocks < 1 ? 1 : blocks);
  };

  if (use_ws) {
    tgn_gather_copy<<<copy_blocks(B), 256, 0, stream>>>(
        (const v4f*)nodeA, source_nodes, (v4f*)(out + o0), (unsigned)B);
    tgn_gather_copy2<<<copy_blocks(B * K), 256, 0, stream>>>(
        (const v4f*)nodeA, neighbors1, (v4f*)(out + o1), (v4f*)(out + o5), (unsigned)(B * K));
    tgn_gather_copy<<<copy_blocks(B * K * K), 256, 0, stream>>>(
        (const v4f*)nodeA, neighbors2, (v4f*)(out + o6), (unsigned)(B * K * K));
  } else {
    tgn_gather_add<<<add_blocks(B), 256, 0, stream>>>(
        (const v4f*)node_feats, (const v4f*)memory, source_nodes,
        (v4f*)(out + o0), nullptr, (unsigned)B);
    tgn_gather_add<<<add_blocks(B * K), 256, 0, stream>>>(
        (const v4f*)node_feats, (const v4f*)memory, neighbors1,
        (v4f*)(out + o1), (v4f*)(out + o5), (unsigned)(B * K));
    tgn_gather_add<<<add_blocks(B * K * K), 256, 0, stream>>>(
        (const v4f*)node_feats, (const v4f*)memory, neighbors2,
        (v4f*)(out + o6), nullptr, (unsigned)(B * K * K));
  }
  // edge gathers (always pure copy)
  tgn_gather_copy<<<copy_blocks(B * K), 256, 0, stream>>>(
      (const v4f*)edge_feats, edge_idxs1, (v4f*)(out + o2), (unsigned)(B * K));
  tgn_gather_copy<<<copy_blocks(B * K * K), 256, 0, stream>>>(
      (const v4f*)edge_feats, edge_idxs2, (v4f*)(out + o7), (unsigned)(B * K * K));

  // deltas + masks
  {
    const unsigned n1 = (unsigned)(B * K);
    tgn_delta_mask_kernel<<<dim3((n1 + 255) / 256), dim3(256), 0, stream>>>(
        timestamps, edge_times1, neighbors1, out + o3, out + o4, n1, (unsigned)K);
    const unsigned n2 = (unsigned)(B * K * K);
    tgn_delta_mask_kernel<<<dim3((n2 + 255) / 256), dim3(256), 0, stream>>>(
        timestamps, edge_times2, neighbors2, out + o8, out + o9, n2, (unsigned)(K * K));
  }
}